// Subtraction2NonCupy_56796647522325
// MI455X (gfx1250) — compile-verified
//
#include <hip/hip_runtime.h>

// Problem constants: input1/input2 = (16, 64, 56, 56) fp32, K=3, pad=1 reflect,
// out = (16, 64, 9, 56*56) fp32.
#define NC_PLANES 1024   // 16*64
#define PLANE_W   56
#define PLANE_HW  3136   // 56*56
#define KK        9
#define BLK       224    // 7 wave32s; 3136/224 == 14 exactly
#define PIX_PER_T 14
#define ASYNC_ITERS 7    // 3136 floats / 2 floats-per-b64 / 224 threads == 7

#if defined(__gfx1250__) && __has_builtin(__builtin_amdgcn_global_load_async_to_lds_b64)
#define HAVE_ASYNC_LDS 1
#else
#define HAVE_ASYNC_LDS 0
#endif

// The async-to-LDS builtin is strictly typed: b64 flavor takes pointers to a
// 2 x i32 vector in global (AS1) and LDS (AS3) address spaces.
typedef int v2i __attribute__((vector_size(2 * sizeof(int))));
typedef __attribute__((address_space(1))) v2i gas_v2i;
typedef __attribute__((address_space(3))) v2i las_v2i;

__global__ __launch_bounds__(BLK)
void sub2_reflect_unfold_kernel(const float* __restrict__ q_in,   // input1
                                const float* __restrict__ x_in,   // input2
                                float* __restrict__ out) {
    __shared__ float tile[PLANE_HW];   // 12.5 KB: one full input2 plane

    const unsigned plane = blockIdx.x;                 // n*64 + c
    const float* __restrict__ src = x_in + (size_t)plane * PLANE_HW;
    const float* __restrict__ qsrc = q_in + (size_t)plane * PLANE_HW;
    float* __restrict__ dst = out + (size_t)plane * (KK * PLANE_HW);

    // ---- Stage the plane into LDS via the CDNA5 async global->LDS path ----
#if HAVE_ASYNC_LDS
    #pragma unroll
    for (int i = 0; i < ASYNC_ITERS; ++i) {
        const unsigned e = (threadIdx.x + i * BLK) * 2u;   // float index, 2 floats / b64
        __builtin_amdgcn_global_load_async_to_lds_b64(
            (gas_v2i*)(src + e),
            (las_v2i*)(&tile[e]),
            /*offset=*/0, /*cpol=*/0);
    }
  #if __has_builtin(__builtin_amdgcn_s_wait_asynccnt)
    __builtin_amdgcn_s_wait_asynccnt(0);
  #else
    asm volatile("s_wait_asynccnt 0" ::: "memory");
  #endif
#else
    // Fallback: plain load + LDS store (still one HBM read of input2).
    #pragma unroll
    for (int i = 0; i < ASYNC_ITERS; ++i) {
        const unsigned e = (threadIdx.x + i * BLK) * 2u;
        tile[e]     = src[e];
        tile[e + 1] = src[e + 1];
    }
#endif
    __syncthreads();

    // ---- Compute: 14 pixels per thread, 9 shifted outputs per pixel ----
    for (int it = 0; it < PIX_PER_T; ++it) {
        const unsigned p  = threadIdx.x + (unsigned)it * BLK;   // 0..3135
        const unsigned oh = p / PLANE_W;
        const unsigned ow = p - oh * PLANE_W;

        const float qv = __builtin_nontemporal_load(qsrc + p);  // streamed, read-once

        #pragma unroll
        for (int di = -1; di <= 1; ++di) {
            int ih = (int)oh + di;
            ih = ih < 0 ? -ih : (ih > PLANE_W - 1 ? 2 * (PLANE_W - 1) - ih : ih); // reflect
            #pragma unroll
            for (int dj = -1; dj <= 1; ++dj) {
                int iw = (int)ow + dj;
                iw = iw < 0 ? -iw : (iw > PLANE_W - 1 ? 2 * (PLANE_W - 1) - iw : iw);
                const int k = (di + 1) * 3 + (dj + 1);
                const float v = tile[ih * PLANE_W + iw];
                // streaming output: non-temporal store, coalesced across lanes
                __builtin_nontemporal_store(qv - v, dst + (size_t)k * PLANE_HW + p);
            }
        }
    }
}

extern "C" void kernel_launch(void* const* d_in, const int* in_sizes, int n_in,
                              void* d_out, int out_size, void* d_ws, size_t ws_size,
                              hipStream_t stream) {
    (void)in_sizes; (void)n_in; (void)out_size; (void)d_ws; (void)ws_size;
    const float* input1 = (const float*)d_in[0];
    const float* input2 = (const float*)d_in[1];
    float* out = (float*)d_out;

    sub2_reflect_unfold_kernel<<<NC_PLANES, BLK, 0, stream>>>(input1, input2, out);
}